// memoryNet_18107582120693
// MI455X (gfx1250) — compile-verified
//
#include <hip/hip_runtime.h>
#include <hip/hip_bf16.h>

typedef __attribute__((ext_vector_type(2))) float v2f;
typedef __attribute__((ext_vector_type(8))) float v8f;

#define FEAT     49152
#define MOD_DIM  12288
#define NMOD     4
#define NMEM     50
#define NMEM_PAD 64
#define EPSF     1e-12f

// workspace layout (float offsets)
#define WS_MASK 0          // 4 floats
#define WS_RNM  4          // 64 floats (rnorm of memory rows, 0 for j>=50)
#define WS_ADDR 128        // B*64 floats (softmax addressing, padded with zeros)

// ---------------------------------------------------------------- kernel 0a
__global__ void k_mask(const int* avail, int n_avail, float* ws) {
    int t = threadIdx.x;
    if (t < NMOD) {
        float m = 0.f;
        for (int i = 0; i < n_avail; ++i)
            if (avail[i] == t) m = 1.f;
        ws[WS_MASK + t] = m;
    }
}

// ---------------------------------------------------------------- kernel 0b
// one block per memory row: rnorm_mem[j] = 1 / max(||memory_j||, eps)
__global__ __launch_bounds__(256) void k_memnorm(const float* __restrict__ mem,
                                                 float* ws) {
    __shared__ float red[256];
    int j = blockIdx.x;
    const float* row = mem + (size_t)j * FEAT;
    float s = 0.f;
    for (int f = threadIdx.x; f < FEAT; f += 256) {
        float v = row[f];
        s += v * v;
    }
    red[threadIdx.x] = s;
    __syncthreads();
    for (int off = 128; off > 0; off >>= 1) {
        if (threadIdx.x < off) red[threadIdx.x] += red[threadIdx.x + off];
        __syncthreads();
    }
    if (threadIdx.x == 0)
        ws[WS_RNM + j] = 1.f / fmaxf(sqrtf(red[0]), EPSF);
}

// ---------------------------------------------------------------- kernel 1
// 16 batch rows per block, 4 waves. sim (WMMA f32 16x16x4) -> softmax -> ws
__global__ __launch_bounds__(128) void k_sim(const float* __restrict__ input,
                                             const float* __restrict__ mem,
                                             float* ws) {
    __shared__ float lds_red[128];
    __shared__ float lds_rnx[16];
    __shared__ float lds_sim[16 * NMEM_PAD];
    __shared__ float lds_mask[NMOD];
    __shared__ float lds_rnm[NMEM_PAD];

    const int t       = threadIdx.x;
    const int rowbase = blockIdx.x * 16;

    if (t < NMOD) lds_mask[t] = ws[WS_MASK + t];
    if (t < NMEM_PAD) lds_rnm[t] = (t < NMEM) ? ws[WS_RNM + t] : 0.f;
    __syncthreads();

    // ---- Phase A: masked sum-of-squares per row (16 rows x 8 sub-lanes)
    {
        int r   = t >> 3;     // 0..15
        int sub = t & 7;      // 0..7
        const float* xrow = input + (size_t)(rowbase + r) * FEAT;
        float s = 0.f;
        for (int m = 0; m < NMOD; ++m) {
            if (lds_mask[m] == 0.f) continue;   // uniform branch
            int base = m * MOD_DIM;
            for (int f = sub; f < MOD_DIM; f += 8) {
                float v = xrow[base + f];
                s += v * v;
            }
        }
        lds_red[t] = s;
    }
    __syncthreads();
    if (t < 16) {
        float acc = 0.f;
        for (int i = 0; i < 8; ++i) acc += lds_red[t * 8 + i];
        lds_rnx[t] = 1.f / fmaxf(sqrtf(acc), EPSF);
    }
    __syncthreads();

    // ---- Phase B: sim tile via V_WMMA_F32_16X16X4_F32
    const int wave = t >> 5;          // 0..3  -> memory cols 16w..16w+15
    const int lane = t & 31;
    const int half = lane >> 4;       // 0/1 -> K pair 0,1 vs 2,3
    const int l16  = lane & 15;
    const int mcol = wave * 16 + l16; // memory row (N), 0..63

    const float* arow   = input + (size_t)(rowbase + l16) * FEAT;
    // Padded lanes (mcol>=50) alias row 49; their accumulator is garbage but
    // finite, and gets multiplied by lds_rnm[mcol]==0 at the LDS write below.
    // No per-element zeroing needed in the hot loop.
    const int    mclamp = (mcol < NMEM) ? mcol : (NMEM - 1);
    const float* brow   = mem + (size_t)mclamp * FEAT;

    v8f acc = {};
    for (int m = 0; m < NMOD; ++m) {
        if (lds_mask[m] == 0.f) continue;       // uniform: skip absent modality
        int base = m * MOD_DIM;
        for (int k = 0; k < MOD_DIM; k += 4) {
            int kk = base + k + 2 * half;
            v2f a; a.x = arow[kk]; a.y = arow[kk + 1];
            v2f b; b.x = brow[kk]; b.y = brow[kk + 1];
            acc = __builtin_amdgcn_wmma_f32_16x16x4_f32(
                false, a, false, b, (short)0, acc, false, false);
        }
    }

    // scatter C/D fragment into LDS with both norm scales applied
    {
        union { v8f v; float f[8]; } u; u.v = acc;
        float cscale = lds_rnm[mcol];           // zero for padded cols
        #pragma unroll
        for (int r8 = 0; r8 < 8; ++r8) {
            int rr = 8 * half + r8;
            lds_sim[rr * NMEM_PAD + mcol] = u.f[r8] * lds_rnx[rr] * cscale;
        }
    }
    __syncthreads();

    // ---- Phase C: per-row softmax over 50 cols, write padded addressing
    if (t < 16) {
        float* srow = lds_sim + t * NMEM_PAD;
        float mx = -1e30f;
        for (int j = 0; j < NMEM; ++j) mx = fmaxf(mx, srow[j]);
        float sum = 0.f;
        for (int j = 0; j < NMEM; ++j) {
            float e = expf(srow[j] - mx);
            srow[j] = e;
            sum += e;
        }
        float inv = 1.f / sum;
        float* dst = ws + WS_ADDR + (size_t)(rowbase + t) * NMEM_PAD;
        for (int j = 0; j < NMEM_PAD; ++j)
            dst[j] = (j < NMEM) ? srow[j] * inv : 0.f;
    }
}

// ---------------------------------------------------------------- kernel 2
// 16 rows x 64 features per block. present modality -> copy; absent -> WMMA
__global__ __launch_bounds__(128) void k_readout(const float* __restrict__ input,
                                                 const float* __restrict__ mem,
                                                 const float* __restrict__ ws,
                                                 float* __restrict__ out) {
    const int rowbase  = blockIdx.y * 16;
    const int fbase    = blockIdx.x * 64;
    const int modality = fbase / MOD_DIM;
    const int t        = threadIdx.x;

    // scalarize the mask so the branch is an SGPR branch (EXEC stays all-1)
    unsigned mbits = __builtin_amdgcn_readfirstlane(
        __float_as_uint(ws[WS_MASK + modality]));
    if (mbits != 0u) {
        // modality present: pass-through copy, coalesced
        for (int e = t; e < 16 * 64; e += 128) {
            int r = e >> 6, c = e & 63;
            size_t idx = (size_t)(rowbase + r) * FEAT + (size_t)(fbase + c);
            out[idx] = input[idx];
        }
        return;
    }

    // modality absent: readout = addressing @ memory via WMMA f32 16x16x4.
    // Addressing is zero for j>=50, so K iterations past 52 contribute
    // nothing and are skipped: 13 WMMA steps cover K=0..51.
    const int wave = t >> 5;
    const int lane = t & 31;
    const int half = lane >> 4;
    const int l16  = lane & 15;
    const int col  = fbase + wave * 16 + l16;   // feature column (N)

    const float* arow = ws + WS_ADDR + (size_t)(rowbase + l16) * NMEM_PAD;

    v8f acc = {};
    #pragma unroll
    for (int k = 0; k < 48; k += 4) {           // kk in [0,47]: no clamp needed
        int kk = k + 2 * half;
        v2f a; a.x = arow[kk]; a.y = arow[kk + 1];
        v2f b;
        b.x = mem[(size_t)kk * FEAT + col];
        b.y = mem[(size_t)(kk + 1) * FEAT + col];
        acc = __builtin_amdgcn_wmma_f32_16x16x4_f32(
            false, a, false, b, (short)0, acc, false, false);
    }
    {   // tail step: kk = 48/49 (half 0) or 50/51 (half 1, a==0, B clamped)
        int kk = 48 + 2 * half;
        v2f a; a.x = arow[kk]; a.y = arow[kk + 1];
        int k0 = (kk     < NMEM) ? kk     : (NMEM - 1);
        int k1 = (kk + 1 < NMEM) ? kk + 1 : (NMEM - 1);
        v2f b;
        b.x = mem[(size_t)k0 * FEAT + col];
        b.y = mem[(size_t)k1 * FEAT + col];
        acc = __builtin_amdgcn_wmma_f32_16x16x4_f32(
            false, a, false, b, (short)0, acc, false, false);
    }

    union { v8f v; float f[8]; } u; u.v = acc;
    #pragma unroll
    for (int r8 = 0; r8 < 8; ++r8) {
        int row = rowbase + 8 * half + r8;
        out[(size_t)row * FEAT + col] = u.f[r8];
    }
}

// ---------------------------------------------------------------- launcher
extern "C" void kernel_launch(void* const* d_in, const int* in_sizes, int n_in,
                              void* d_out, int out_size, void* d_ws, size_t ws_size,
                              hipStream_t stream) {
    (void)n_in; (void)out_size; (void)ws_size;

    const float* input = (const float*)d_in[0];
    const float* mem   = (const float*)d_in[1];
    const int*   avail = (const int*)d_in[2];
    const int    n_avail = in_sizes[2];
    const int    Brows   = in_sizes[0] / FEAT;   // 2048

    float* ws  = (float*)d_ws;
    float* out = (float*)d_out;

    k_mask<<<1, 32, 0, stream>>>(avail, n_avail, ws);
    k_memnorm<<<NMEM, 256, 0, stream>>>(mem, ws);
    k_sim<<<Brows / 16, 128, 0, stream>>>(input, mem, ws);

    dim3 g2(FEAT / 64, Brows / 16);
    k_readout<<<g2, 128, 0, stream>>>(input, mem, ws, out);
}